// EmbeddingNet_87205015978174
// MI455X (gfx1250) — compile-verified
//
#include <hip/hip_runtime.h>
#include <hip/hip_bf16.h>

// Problem constants (from the reference)
#define BATCH   2048
#define SEQ     201
#define HALF    100            // SEQ / 2
#define STACKN  101            // HALF + 1
#define EMB     128
#define M_ROWS  (BATCH * SEQ)  // 411648 = 16 * 25728

typedef __attribute__((ext_vector_type(2))) float v2f;
typedef __attribute__((ext_vector_type(4))) float v4f;
typedef __attribute__((ext_vector_type(8))) float v8f;

// ---------------------------------------------------------------------------
// Kernel 1: sequential pointer-chase per batch element.
// One thread = one batch row. Stack lives in LDS, per-thread stride 101
// (odd stride -> all 32 lanes hit distinct LDS banks on each scan step).
// Writes visited_time (i32 scratch + i64 output) and top2 (i32 output).
// ---------------------------------------------------------------------------
__global__ __launch_bounds__(128) void chase_kernel(
    const long long* __restrict__ sol,   // [BATCH][SEQ] int64
    int*             __restrict__ vt32,  // [BATCH][SEQ] scratch (ws)
    long long*       __restrict__ vt64,  // [BATCH][SEQ] output
    int*             __restrict__ top2)  // [BATCH][SEQ][2] output
{
    __shared__ float stack_lds[128 * STACKN];   // 51.7 KB of the 320 KB WGP LDS
    const int b = blockIdx.x * 128 + threadIdx.x;
    float* st = &stack_lds[threadIdx.x * STACKN];

    for (int j = 0; j < STACKN; ++j) st[j] = -0.01f;
    st[0] = 0.0f;

    const long long* srow = sol + (size_t)b * SEQ;
    int pre = 0;
    for (int i = 0; i < SEQ; ++i) {
        const int cur = (int)srow[pre];
        const int t   = i + 1;
        const size_t o = (size_t)b * SEQ + cur;
        vt32[o] = t;
        vt64[o] = (long long)t;

        // push for pickups, pop (-0.01) for deliveries, no-op for depot
        const int   pos = (cur > HALF) ? (cur - HALF) : cur;
        const float val = (cur == 0) ? st[0]
                        : (cur > HALF) ? -0.01f
                                       : (float)t;
        st[pos] = val;

        // top-2 scan of the 101-entry stack (ties -> lowest index, like top_k)
        float b1 = -3.0e38f, b2 = -3.0e38f;
        int   i1 = 0,        i2 = 0;
#pragma unroll 1
        for (int j = 0; j < STACKN; ++j) {
            const float v = st[j];
            if (v > b1)      { b2 = b1; i2 = i1; b1 = v; i1 = j; }
            else if (v > b2) { b2 = v;  i2 = j; }
        }
        top2[o * 2 + 0] = i1;
        top2[o * 2 + 1] = i2;
        pre = cur;
    }
}

// ---------------------------------------------------------------------------
// Kernel 2: pos_enc[b,s,:] = pattern[visited_time[b,s] % SEQ, :]
// Stage the whole pattern (102,912 B, legal: up to 320 KB LDS per workgroup)
// into LDS with gfx1250 async copies, then gather from LDS and stream the
// 211 MB of output with non-temporal stores (output >> 192 MB L2, so NT
// avoids thrashing L2 while keeping vt32 reads cached).
// One lane per float4: a row (128 floats) = exactly one wave32 of float4s,
// and 32 consecutive float4 LDS reads are bank-conflict-free.
// ---------------------------------------------------------------------------
extern __shared__ float lds_pat[];   // SEQ*EMB floats (dynamic LDS)

__global__ __launch_bounds__(256) void posenc_kernel(
    const float* __restrict__ pattern,   // [SEQ][EMB]
    const int*   __restrict__ vt32,      // [M_ROWS]
    float*       __restrict__ pos_out)   // [M_ROWS][EMB]
{
    // ---- async global -> LDS staging of the pattern table ----
    const int NP4 = SEQ * EMB / 4;   // 6432 16-byte transfers
    const unsigned lds0 = (unsigned)(unsigned long long)(uintptr_t)lds_pat;
    for (int i = threadIdx.x; i < NP4; i += 256) {
        const unsigned goff = (unsigned)i * 16u;        // byte offset in pattern
        const unsigned loff = lds0 + (unsigned)i * 16u; // LDS byte address
        asm volatile("global_load_async_to_lds_b128 %0, %1, %2"
                     :: "v"(loff), "v"(goff), "s"(pattern)
                     : "memory");
    }
    asm volatile("s_wait_asynccnt 0x0" ::: "memory");
    __syncthreads();   // other waves staged other chunks

    const v4f* lp4  = (const v4f*)lds_pat;
    v4f*       out4 = (v4f*)pos_out;
    const size_t total = (size_t)M_ROWS * 32;   // float4s
    for (size_t g = (size_t)blockIdx.x * blockDim.x + threadIdx.x;
         g < total;
         g += (size_t)gridDim.x * blockDim.x) {
        const size_t r   = g >> 5;         // row
        const int    c   = (int)(g & 31);  // float4 column
        const int    idx = vt32[r] % SEQ;
        const v4f    val = lp4[(size_t)idx * 32 + c];
        __builtin_nontemporal_store(val, &out4[r * 32 + c]);
    }
}

// ---------------------------------------------------------------------------
// Kernel 3: x_embedding = X[M,2] @ W^T[2,128] via V_WMMA_F32_16X16X4_F32.
// A layout (32-bit 16x4): lanes 0-15 carry M=lane, VGPR0=K0, VGPR1=K1;
// lanes 16-31 carry K=2,3 -> zero (K is only 2). B layout mirrors it.
// Loads are unconditional (all addresses in-bounds via lane&15), zeros are
// applied with selects afterwards -> v_cndmask instead of EXEC branches.
// One wave per 16x16 output tile; 8 waves/block = all 8 N-tiles of an M-tile.
// ---------------------------------------------------------------------------
__global__ __launch_bounds__(256) void emb_wmma_kernel(
    const float* __restrict__ x,    // [M_ROWS][2]
    const float* __restrict__ W,    // [EMB][2]
    float*       __restrict__ out)  // [M_ROWS][EMB]
{
    const int wave  = threadIdx.x >> 5;   // N-tile 0..7
    const int lane  = threadIdx.x & 31;
    const int l16   = lane & 15;
    const bool lo   = lane < 16;
    const int mtile = blockIdx.x;         // 0..25727

    const size_t row = (size_t)mtile * 16 + l16;
    const int    col = wave * 16 + l16;

    // unconditional loads (in-bounds for every lane), then select-to-zero
    const float a0 = x[row * 2 + 0];
    const float a1 = x[row * 2 + 1];
    const float w0 = W[col * 2 + 0];
    const float w1 = W[col * 2 + 1];

    v2f a, bm;
    a.x  = lo ? a0 : 0.0f;    // K=0 (upper half lanes: K=2 -> 0)
    a.y  = lo ? a1 : 0.0f;    // K=1 (upper half lanes: K=3 -> 0)
    bm.x = lo ? w0 : 0.0f;    // B row K=0
    bm.y = lo ? w1 : 0.0f;    // B row K=1

    v8f c = {};
    // (neg_a, A, neg_b, B, c_mod, C, reuse_a, reuse_b) -> v_wmma_f32_16x16x4_f32
    c = __builtin_amdgcn_wmma_f32_16x16x4_f32(false, a, false, bm,
                                              (short)0, c, false, false);

    // D layout: VGPR v, lanes 0-15 -> M = v, lanes 16-31 -> M = v + 8
    const size_t rbase = (size_t)mtile * 16 + (lo ? 0 : 8);
#pragma unroll
    for (int v = 0; v < 8; ++v) {
        __builtin_nontemporal_store(c[v], &out[(rbase + v) * EMB + col]);
    }
}

// ---------------------------------------------------------------------------
// Launch: d_out layout = [x_embedding f32 | pos_enc f32 | visited_time i64 |
// top2 i32], flat in reference return order.
// ---------------------------------------------------------------------------
extern "C" void kernel_launch(void* const* d_in, const int* in_sizes, int n_in,
                              void* d_out, int out_size, void* d_ws, size_t ws_size,
                              hipStream_t stream) {
    const float*     x       = (const float*)d_in[0];      // [B,S,2] f32
    const long long* sol     = (const long long*)d_in[1];  // [B,S]   i64
    const float*     W       = (const float*)d_in[2];      // [128,2] f32
    const float*     pattern = (const float*)d_in[3];      // [S,128] f32

    float* out    = (float*)d_out;
    float* x_emb  = out;                                   // M_ROWS*EMB f32
    float* posenc = out + (size_t)M_ROWS * EMB;            // M_ROWS*EMB f32
    long long* vt64 = (long long*)(out + 2 * (size_t)M_ROWS * EMB);
    int*  top2  = (int*)(out + 2 * (size_t)M_ROWS * EMB + 2 * (size_t)BATCH * SEQ);
    int*  vt32  = (int*)d_ws;                              // scratch: M_ROWS i32

    // 1) sequential chase (produces vt32 needed by the gather)
    chase_kernel<<<BATCH / 128, 128, 0, stream>>>(sol, vt32, vt64, top2);
    // 2) WMMA embedding GEMM (independent of the chase)
    emb_wmma_kernel<<<M_ROWS / 16, 256, 0, stream>>>(x, W, x_emb);
    // 3) positional-encoding gather (same stream -> ordered after chase)
    posenc_kernel<<<512, 256, SEQ * EMB * sizeof(float), stream>>>(pattern, vt32, posenc);
}